// CodeAwareContinuousEncoder_11622181503255
// MI455X (gfx1250) — compile-verified
//
#include <hip/hip_runtime.h>
#include <math.h>

#define B_N    4096
#define H_N    256
#define E_N    256
#define K_N    100
#define TILE_M 32
#define KP     32            // k-panel rows staged in LDS per step
#define WPITCH 544           // floats per kpair row in s_w2: 512 + 32 pad (bank shift)
#define HPITCH 96            // floats per hpair row in s_h2: 64 + 32 pad (bank shift)

typedef __attribute__((ext_vector_type(2))) float v2f;
typedef __attribute__((ext_vector_type(8))) float v8f;

// ---------------- bucketing kernels ----------------

__global__ void zero_counts(int* counts, int* cursor) {
    counts[threadIdx.x] = 0;
    cursor[threadIdx.x] = 0;
}

__global__ void count_heads(const int* __restrict__ head_idx, int* counts) {
    int b = blockIdx.x * blockDim.x + threadIdx.x;
    if (b < B_N) atomicAdd(&counts[head_idx[b]], 1);
}

__global__ void scan_heads(const int* __restrict__ counts, int* offsets, int* cursor) {
    if (threadIdx.x == 0) {
        int run = 0;
        for (int k = 0; k < K_N; ++k) {
            offsets[k] = run;
            cursor[k]  = run;
            run += counts[k];
        }
        offsets[K_N] = run;
    }
}

__global__ void scatter_heads(const int* __restrict__ head_idx, int* cursor, int* order) {
    int b = blockIdx.x * blockDim.x + threadIdx.x;
    if (b < B_N) {
        int p = atomicAdd(&cursor[head_idx[b]], 1);
        order[p] = b;
    }
}

// ---------------- per-head GEMM with f32 WMMA ----------------
// Block = 256 threads = 8 wave32. One block: one (head, 32-sample tile).
// M=32, N=E=256 (32 cols/wave), K=H=256. Per wave: 4 accumulators
// (2 M-subtiles x 2 N-subtiles), 4 WMMAs per k-step from 4 b64 LDS loads.
__global__ __launch_bounds__(256)
void head_gemm(const float* __restrict__ values, const float* __restrict__ means,
               const float* __restrict__ stds,
               const float* __restrict__ w1, const float* __restrict__ b1,
               const float* __restrict__ W_heads, const float* __restrict__ b_heads,
               const int* __restrict__ offsets, const int* __restrict__ order,
               float* __restrict__ out) {
    // pair-interleaved: s_h2[hpair][m][{h,h+1}], s_w2[kpair][n][{k,k+1}]
    __shared__ __align__(16) float s_h2[(H_N / 2) * HPITCH];   // 48 KB
    __shared__ __align__(16) float s_w2[(KP / 2) * WPITCH];    // 34 KB
    __shared__ float s_z[TILE_M];
    __shared__ int   s_sid[TILE_M];

    const int head  = blockIdx.y;
    const int tile  = blockIdx.x;
    const int start = offsets[head];
    const int cnt   = offsets[head + 1] - start;
    if (tile * TILE_M >= cnt) return;   // uniform early-exit (EXEC all-1 for WMMA)

    const int tid = threadIdx.x;

    // ---- per-sample scalar normalization for this tile's 32 samples ----
    if (tid < TILE_M) {
        int r   = tile * TILE_M + tid;
        int sid = (r < cnt) ? order[start + r] : -1;
        s_sid[tid] = sid;
        float z = 0.f;
        if (sid >= 0) {
            float s = fmaxf(stds[sid], 1e-8f);
            z = (values[sid] - means[sid]) / s;
            z = fminf(fmaxf(z, -5.f), 5.f);
        }
        s_z[tid] = z;
    }
    __syncthreads();

    // ---- hidden tile on the fly (exact GELU), stored pair-interleaved ----
    {
        const int   h  = tid;                  // 256 threads == H_N columns
        const float w  = w1[h];
        const float bb = b1[h];
        float* dst = &s_h2[(h >> 1) * HPITCH + (h & 1)];
        #pragma unroll 4
        for (int m = 0; m < TILE_M; ++m) {
            float v = 0.f;
            if (s_sid[m] >= 0) {
                float x = fmaf(s_z[m], w, bb);
                v = 0.5f * x * (1.f + erff(x * 0.70710678118654752f));
            }
            dst[m * 2] = v;                    // zero rows beyond cnt
        }
    }

    const int wave  = tid >> 5;
    const int lane  = tid & 31;
    const int n0    = wave * 32;               // 32 output cols per wave
    const int ln15  = lane & 15;               // M (A) / N-within-tile (B)
    const int hiHalf = lane >> 4;              // lanes 16-31 hold K+2/K+3

    v8f c00 = {0.f,0.f,0.f,0.f,0.f,0.f,0.f,0.f};
    v8f c01 = c00, c10 = c00, c11 = c00;

    const float* Wk = W_heads + (size_t)head * (H_N * E_N);

    for (int kp = 0; kp < H_N; kp += KP) {
        __syncthreads();                       // prior panel consumed
        // stage panel rows [kp, kp+KP): b128 global loads of row-pairs,
        // shuffled into pair-interleaved b128 LDS stores
        #pragma unroll
        for (int j = 0; j < 4; ++j) {
            int c  = j * 256 + tid;            // chunk: 16 kpairs x 64 col-quads
            int r  = c >> 6;                   // kpair within panel
            int n4 = (c & 63) * 4;
            const float4 va = *(const float4*)(Wk + (size_t)(kp + 2 * r)     * E_N + n4);
            const float4 vb = *(const float4*)(Wk + (size_t)(kp + 2 * r + 1) * E_N + n4);
            float4* d = (float4*)&s_w2[r * WPITCH + n4 * 2];
            d[0] = make_float4(va.x, vb.x, va.y, vb.y);
            d[1] = make_float4(va.z, vb.z, va.w, vb.w);
        }
        if (kp + KP < H_N)
            __builtin_prefetch((const void*)(Wk + (size_t)(kp + KP) * E_N + tid * 32), 0, 0);
        __syncthreads();

        #pragma unroll
        for (int ks = 0; ks < KP; ks += 4) {
            // A frags: hpair row, lanes 16-31 take hpair+1 (K+2/K+3)
            const int hp = ((kp + ks) >> 1) + hiHalf;
            const float* ah = &s_h2[hp * HPITCH + ln15 * 2];
            v2f a0 = *(const v2f*)(ah);                // M-subtile 0 (m 0..15)
            v2f a1 = *(const v2f*)(ah + 32);           // M-subtile 1 (m 16..31)
            // B frags: kpair row within panel
            const int rp = (ks >> 1) + hiHalf;
            const float* bh = &s_w2[rp * WPITCH + (n0 + ln15) * 2];
            v2f b0 = *(const v2f*)(bh);                // N-subtile 0
            v2f b1 = *(const v2f*)(bh + 32);           // N-subtile 1

            c00 = __builtin_amdgcn_wmma_f32_16x16x4_f32(false, a0, false, b0, (short)0, c00, false, false);
            c01 = __builtin_amdgcn_wmma_f32_16x16x4_f32(false, a0, false, b1, (short)0, c01, false, false);
            c10 = __builtin_amdgcn_wmma_f32_16x16x4_f32(false, a1, false, b0, (short)0, c10, false, false);
            c11 = __builtin_amdgcn_wmma_f32_16x16x4_f32(false, a1, false, b1, (short)0, c11, false, false);
        }
    }

    // ---- epilogue: C layout VGPR r -> M = r + 8*(lane>=16), N = ln15 ----
    const float* bg  = b_heads + head * E_N;
    const float  bgn0 = bg[n0 + ln15];
    const float  bgn1 = bg[n0 + 16 + ln15];
    const int    mb  = hiHalf << 3;
    #pragma unroll
    for (int r = 0; r < 8; ++r) {
        int m0  = r + mb;
        int m1  = 16 + r + mb;
        int sid0 = s_sid[m0];
        int sid1 = s_sid[m1];
        if (sid0 >= 0) {
            out[(size_t)sid0 * E_N + n0 + ln15]      = c00[r] + bgn0;
            out[(size_t)sid0 * E_N + n0 + 16 + ln15] = c01[r] + bgn1;
        }
        if (sid1 >= 0) {
            out[(size_t)sid1 * E_N + n0 + ln15]      = c10[r] + bgn0;
            out[(size_t)sid1 * E_N + n0 + 16 + ln15] = c11[r] + bgn1;
        }
    }
}

// ---------------- launcher ----------------

extern "C" void kernel_launch(void* const* d_in, const int* in_sizes, int n_in,
                              void* d_out, int out_size, void* d_ws, size_t ws_size,
                              hipStream_t stream) {
    const float* values   = (const float*)d_in[0];
    const float* means    = (const float*)d_in[1];
    const float* stds     = (const float*)d_in[2];
    const int*   head_idx = (const int*)d_in[3];
    const float* w1       = (const float*)d_in[4];
    const float* b1       = (const float*)d_in[5];
    const float* W_heads  = (const float*)d_in[6];
    const float* b_heads  = (const float*)d_in[7];
    float*       out      = (float*)d_out;

    // workspace: counts[128] | offsets[160] | cursor[128] | order[B]  (~18 KB)
    int* counts  = (int*)d_ws;
    int* offsets = counts + 128;
    int* cursor  = offsets + 160;
    int* order   = cursor + 128;

    zero_counts  <<<1, 128, 0, stream>>>(counts, cursor);
    count_heads  <<<B_N / 256, 256, 0, stream>>>(head_idx, counts);
    scan_heads   <<<1, 32, 0, stream>>>(counts, offsets, cursor);
    scatter_heads<<<B_N / 256, 256, 0, stream>>>(head_idx, cursor, order);

    dim3 grid(B_N / TILE_M, K_N);   // worst-case tiles per head; excess blocks exit early
    head_gemm<<<grid, 256, 0, stream>>>(values, means, stds, w1, b1,
                                        W_heads, b_heads, offsets, order, out);
}